// ColorTransforms_1297080123783
// MI455X (gfx1250) — compile-verified
//
#include <hip/hip_runtime.h>
#include <hip/hip_bf16.h>

// Problem constants (from reference): imgs [32,3,224,224] f32, LUT [32,72,72,72,3] f32
#define RL      72
#define HW      50176            // 224*224
#define NIMG    32
#define TOTPIX  (NIMG * HW)      // 1,605,632
#define CHUNK   256
#define NCHUNK  (TOTPIX / CHUNK) // 6272 exactly
#define NBLK    1568             // 6272 / 1568 = 4 chunks per block exactly
#define LUTF    (RL * RL * RL * 3) // floats per image LUT = 1,119,744
#define SX      (RL * RL * 3)    // 15552  (x stride in floats)
#define SY      (RL * 3)         // 216    (y stride in floats)

struct F3 { float x, y, z; };

__device__ __forceinline__ F3 ld3(const float* __restrict__ p) {
    F3 v; v.x = p[0]; v.y = p[1]; v.z = p[2]; return v;
}
__device__ __forceinline__ F3 lp3(F3 a, F3 b, float t) {
    F3 r;
    r.x = a.x * (1.0f - t) + b.x * t;
    r.y = a.y * (1.0f - t) + b.y * t;
    r.z = a.z * (1.0f - t) + b.z * t;
    return r;
}

// CDNA5 async copy: global -> LDS, per-lane b32, tracked by ASYNCcnt.
// %0 = 32-bit LDS byte address VGPR, %1 = 64-bit global address VGPR pair.
__device__ __forceinline__ void async_g2l_b32(unsigned lds_addr, const float* gsrc) {
    asm volatile("global_load_async_to_lds_b32 %0, %1, off"
                 :: "v"(lds_addr), "v"((unsigned long long)gsrc)
                 : "memory");
}
__device__ __forceinline__ void wait_async0() {
    asm volatile("s_wait_asynccnt 0" ::: "memory");
}
__device__ __forceinline__ void wait_ds0() {
    asm volatile("s_wait_dscnt 0" ::: "memory");
}

// Low 32 bits of a generic pointer to __shared__ == LDS byte offset (ISA §10.2 aperture rule).
__device__ __forceinline__ unsigned lds_off(const void* p) {
    return (unsigned)(unsigned long long)p;
}

__global__ __launch_bounds__(256)
void trilut_kernel(const float* __restrict__ imgs,
                   const float* __restrict__ lut,
                   float* __restrict__ out) {
    __shared__ float sbuf[2][3][CHUNK];   // double-buffered staged RGB
    const int tid = threadIdx.x;

    // Issue async loads of one chunk's RGB into LDS buffer `buf`.
    // Each lane fetches only its own 3 channel values -> no inter-wave hazard.
    auto issue = [&](int chunk, int buf) {
        const int p   = chunk * CHUNK + tid;      // always < TOTPIX (exact geometry)
        const int n   = p / HW;
        const int rem = p - n * HW;
        const float* base = imgs + (size_t)n * 3 * HW + rem;
        async_g2l_b32(lds_off(&sbuf[buf][0][tid]), base);
        async_g2l_b32(lds_off(&sbuf[buf][1][tid]), base + HW);
        async_g2l_b32(lds_off(&sbuf[buf][2][tid]), base + 2 * HW);
    };

    int c   = blockIdx.x;
    int buf = 0;
    if (c < NCHUNK) issue(c, 0);   // prologue

    for (; c < NCHUNK; c += gridDim.x) {
        wait_async0();                               // current buffer's data has landed
        float r = sbuf[buf][0][tid];
        float g = sbuf[buf][1][tid];
        float b = sbuf[buf][2][tid];
        wait_ds0();                                  // LDS reads retired before next overwrite

        const int nc = c + gridDim.x;                // block-uniform
        if (nc < NCHUNK) issue(nc, buf ^ 1);         // overlap next input with gathers below

        const int p   = c * CHUNK + tid;
        const int n   = p / HW;
        const int rem = p - n * HW;
        const float* __restrict__ L = lut + (size_t)n * LUTF;

        // scale to [0,71], split integer / fractional parts (matches jnp floor + mod 1)
        const float sx = r * 71.0f, sy = g * 71.0f, sz = b * 71.0f;
        const float wx = floorf(sx), wy = floorf(sy), wz = floorf(sz);
        const float fx = sx - wx,   fy = sy - wy,    fz = sz - wz;
        int ix = (int)wx, iy = (int)wy, iz = (int)wz;

        const int x0 = min(max(ix, 0), RL - 1), x1 = min(max(ix + 1, 0), RL - 1);
        const int y0 = min(max(iy, 0), RL - 1), y1 = min(max(iy + 1, 0), RL - 1);
        const int z0 = min(max(iz, 0), RL - 1), z1 = min(max(iz + 1, 0), RL - 1);

        const int ax0 = x0 * SX, ax1 = x1 * SX;
        const int ay0 = y0 * SY, ay1 = y1 * SY;
        const int az0 = z0 * 3,  az1 = z1 * 3;

        // 8 independent 12B gathers: issued together, served from L2-resident LUT
        F3 c000 = ld3(L + ax0 + ay0 + az0);
        F3 c001 = ld3(L + ax0 + ay0 + az1);
        F3 c010 = ld3(L + ax0 + ay1 + az0);
        F3 c011 = ld3(L + ax0 + ay1 + az1);
        F3 c100 = ld3(L + ax1 + ay0 + az0);
        F3 c101 = ld3(L + ax1 + ay0 + az1);
        F3 c110 = ld3(L + ax1 + ay1 + az0);
        F3 c111 = ld3(L + ax1 + ay1 + az1);

        F3 c00 = lp3(c000, c001, fz);
        F3 c01 = lp3(c010, c011, fz);
        F3 c10 = lp3(c100, c101, fz);
        F3 c11 = lp3(c110, c111, fz);
        F3 c0  = lp3(c00, c01, fy);
        F3 c1  = lp3(c10, c11, fy);
        F3 cc  = lp3(c0, c1, fx);

        float ro = fminf(fmaxf(cc.x, 0.0f), 1.0f);
        float go = fminf(fmaxf(cc.y, 0.0f), 1.0f);
        float bo = fminf(fmaxf(cc.z, 0.0f), 1.0f);

        // Non-temporal streaming stores: don't evict the L2-resident LUTs
        float* o = out + (size_t)n * 3 * HW + rem;
        __builtin_nontemporal_store(ro, o);
        __builtin_nontemporal_store(go, o + HW);
        __builtin_nontemporal_store(bo, o + 2 * HW);

        buf ^= 1;
    }
}

extern "C" void kernel_launch(void* const* d_in, const int* in_sizes, int n_in,
                              void* d_out, int out_size, void* d_ws, size_t ws_size,
                              hipStream_t stream) {
    const float* imgs = (const float*)d_in[0];   // [32,3,224,224]
    const float* lut  = (const float*)d_in[1];   // [32,72,72,72,3]
    float* out        = (float*)d_out;           // [32,3,224,224]
    (void)in_sizes; (void)n_in; (void)out_size; (void)d_ws; (void)ws_size;

    trilut_kernel<<<NBLK, CHUNK, 0, stream>>>(imgs, lut, out);
}